// YOLOv9AddNMS_50775103373485
// MI455X (gfx1250) — compile-verified
//
#include <hip/hip_runtime.h>
#include <stdint.h>

#define NANCH   8400
#define NCH     85
#define NCLS    80
#define KTOP    1024
#define NSORT   16384
#define MAXDET  100
#define CONF_T  0.25f
#define IOU_T   0.45f
#define MAX_WH  7680.0f

typedef float v2f __attribute__((ext_vector_type(2)));
typedef float v8f __attribute__((ext_vector_type(8)));

// ---------------- LDS layout (bytes) -------------------------------------
// Region R is time-shared: bitonic sort keys (u64[16384] = 128KB) first,
// then the 1024x1024 IoU bit matrix (u32[1024*32] = 128KB).
#define OFF_R      0
#define OFF_OBOX   (131072)                 // float[1024*4]  offset boxes
#define OFF_RBOX   (OFF_OBOX  + 16384)      // float[1024*4]  raw xyxy boxes
#define OFF_AREA   (OFF_RBOX  + 16384)      // float[1024]
#define OFF_SCORE  (OFF_AREA  + 4096)       // float[1024]
#define OFF_CLS    (OFF_SCORE + 4096)       // float[1024]
#define OFF_CIDX   (OFF_CLS   + 4096)       // u8[8448] per-anchor argmax class
#define OFF_KEEP   (OFF_CIDX  + 8448)       // u32[32]
#define OFF_PREF   (OFF_KEEP  + 128)        // u32[32]
#define SMEM_BYTES (OFF_PREF  + 128)        // 184832 B < 320KB WGP LDS

__device__ __forceinline__ uint32_t f2ord(float f) {
  uint32_t u = __float_as_uint(f);
  return (u & 0x80000000u) ? ~u : (u | 0x80000000u);   // ascending order key
}
__device__ __forceinline__ float ord2f(uint32_t o) {
  uint32_t u = (o & 0x80000000u) ? (o & 0x7FFFFFFFu) : ~o;
  return __uint_as_float(u);
}

__global__ __launch_bounds__(1024)
void yolo_nms_kernel(const float* __restrict__ pred, float* __restrict__ out) {
  extern __shared__ unsigned char smem[];
  uint64_t* keys  = (uint64_t*)(smem + OFF_R);
  uint32_t* ioub  = (uint32_t*)(smem + OFF_R);
  float*    obox  = (float*)   (smem + OFF_OBOX);
  float*    rbox  = (float*)   (smem + OFF_RBOX);
  float*    area  = (float*)   (smem + OFF_AREA);
  float*    score = (float*)   (smem + OFF_SCORE);
  float*    clsf  = (float*)   (smem + OFF_CLS);
  uint8_t*  cidx  = (uint8_t*) (smem + OFF_CIDX);
  uint32_t* keepw = (uint32_t*)(smem + OFF_KEEP);
  uint32_t* pref  = (uint32_t*)(smem + OFF_PREF);

  const int tid  = threadIdx.x;
  const int lane = tid & 31;
  const int wave = tid >> 5;
  const int b    = blockIdx.x;
  const float* P = pred + (size_t)b * NANCH * NCH;

  // ---- Phase A: per-anchor confidence / class argmax -> sort keys --------
  for (int a = tid; a < NANCH; a += 1024) {
    const float* r = P + (size_t)a * NCH;
    __builtin_prefetch(r + NCH * 1024, 0, 1);   // global_prefetch_b8
    float obj  = r[4];
    float best = r[5];
    int   bj   = 0;
    #pragma unroll 4
    for (int c = 1; c < NCLS; ++c) {
      float v = r[5 + c];
      if (v > best) { best = v; bj = c; }       // first-max (matches argmax)
    }
    float conf  = obj * best;
    bool  valid = (obj > CONF_T) && (conf > CONF_T);
    float s     = valid ? conf : -1.0f;
    // descending score, stable by ascending index under ascending u64 sort
    keys[a] = ((uint64_t)(~f2ord(s)) << 32) | (uint32_t)a;
    cidx[a] = (uint8_t)bj;
  }
  for (int a = NANCH + tid; a < NSORT; a += 1024) keys[a] = ~0ull;

  // ---- Phase B: bitonic sort of 16384 keys (ascending) -------------------
  for (int k = 2; k <= NSORT; k <<= 1) {
    for (int j = k >> 1; j > 0; j >>= 1) {
      __syncthreads();
      for (int i = tid; i < NSORT; i += 1024) {
        int ixj = i ^ j;
        if (ixj > i) {
          uint64_t x = keys[i], y = keys[ixj];
          bool up = ((i & k) == 0);
          if ((x > y) == up) { keys[i] = y; keys[ixj] = x; }
        }
      }
    }
  }
  __syncthreads();

  // ---- Gather top-K candidate state; then recycle region R ---------------
  uint64_t mykey = (tid < KTOP) ? keys[tid] : ~0ull;
  __syncthreads();                       // everyone done reading keys
  for (int i = tid; i < KTOP * 32; i += 1024) ioub[i] = 0;   // reuse R
  if (tid < KTOP) {
    uint32_t idx = (uint32_t)mykey;
    float    s   = ord2f(~(uint32_t)(mykey >> 32));
    if (idx < NANCH) {
      const float* r = P + (size_t)idx * NCH;
      float cx = r[0], cy = r[1], w = r[2], h = r[3];
      float x1 = cx - 0.5f * w, y1 = cy - 0.5f * h;
      float x2 = cx + 0.5f * w, y2 = cy + 0.5f * h;
      float cf  = (float)cidx[idx];
      float off = cf * MAX_WH;
      rbox[tid*4+0] = x1;       rbox[tid*4+1] = y1;
      rbox[tid*4+2] = x2;       rbox[tid*4+3] = y2;
      obox[tid*4+0] = x1 + off; obox[tid*4+1] = y1 + off;
      obox[tid*4+2] = x2 + off; obox[tid*4+3] = y2 + off;
      area[tid]  = (x2 - x1) * (y2 - y1);
      score[tid] = s;
      clsf[tid]  = cf;
    } else {                               // padding candidate: never kept
      float base = -1.0e7f - (float)tid * 4.0f;
      rbox[tid*4+0] = 0.f; rbox[tid*4+1] = 0.f; rbox[tid*4+2] = 0.f; rbox[tid*4+3] = 0.f;
      obox[tid*4+0] = base;        obox[tid*4+1] = base;
      obox[tid*4+2] = base + 1.f;  obox[tid*4+3] = base + 1.f;
      area[tid]  = 1.0f;
      score[tid] = -2.0f;
      clsf[tid]  = 0.0f;
    }
  }
  __syncthreads();

  // ---- Phase C: 1024x1024 IoU>thr bit matrix -----------------------------
  // Wave w owns 16-row tiles I = 2w, 2w+1 (rows [32w, 32w+32)) -> race-free
  // bit packing. Per 16x16 tile: area_i only via V_WMMA_F32_16X16X4_F32 (area_j sits at K=2 where A is 0)
  // (A row=[area,1,0,0], B col=[1,0,area,0]); intersections via VALU.
  {
    const float4* ob4 = (const float4*)obox;
    for (int Ii = 0; Ii < 2; ++Ii) {
      const int I = wave * 2 + Ii;
#if __has_builtin(__builtin_amdgcn_wmma_f32_16x16x4_f32)
      v2f afr;   // A 16x4: lanes 0-15 hold K=0,1 ; lanes 16-31 K=2,3 (zero)
      afr.x = (lane < 16) ? area[I*16 + lane] : 0.0f;
      afr.y = (lane < 16) ? 1.0f              : 0.0f;
#endif
      for (int J = 0; J < 64; ++J) {
        float sums[8];
#if __has_builtin(__builtin_amdgcn_wmma_f32_16x16x4_f32)
        v2f bfr;  // B 4x16: K=0 row = ones (lanes 0-15), K=2 row = area_b (lanes 16-31), K=1,3 rows zero
        bfr.x = (lane < 16) ? 1.0f : area[J*16 + (lane - 16)];
        bfr.y = 0.0f;
        v8f c = {0.f,0.f,0.f,0.f,0.f,0.f,0.f,0.f};
        c = __builtin_amdgcn_wmma_f32_16x16x4_f32(
                false, afr, false, bfr, (short)0, c, false, false);
        #pragma unroll
        for (int k = 0; k < 8; ++k) sums[k] = c[k];
#else
        const int   ibase = I*16 + ((lane >= 16) ? 8 : 0);
        const float aj    = area[J*16 + (lane & 15)];
        #pragma unroll
        for (int k = 0; k < 8; ++k) sums[k] = area[ibase + k] + aj;
#endif
        const int jcol = J*16 + (lane & 15);   // C layout: col = lane%16
        float4 bb = ob4[jcol];
        #pragma unroll
        for (int k = 0; k < 8; ++k) {
          int irow = I*16 + k + ((lane >= 16) ? 8 : 0);   // C layout row
          float4 aa = ob4[irow];
          float wI = fminf(aa.z, bb.z) - fmaxf(aa.x, bb.x);
          float hI = fminf(aa.w, bb.w) - fmaxf(aa.y, bb.y);
          float inter = fmaxf(wI, 0.0f) * fmaxf(hI, 0.0f);
          // inter/(sum - inter + eps) > T  <=>  inter*(1+T) > T*(sum+eps)
          bool cond = (inter * (1.0f + IOU_T)) > (IOU_T * (sums[k] + 1e-7f));
          uint32_t m = (uint32_t)__ballot(cond);   // wave32 ballot
          if (lane == 0) {
            int word = J >> 1, sh = (J & 1) * 16;
            ioub[(I*16 + k    )*32 + word] |= (m & 0xFFFFu) << sh;
            ioub[(I*16 + k + 8)*32 + word] |= (m >> 16)     << sh;
          }
        }
      }
    }
  }
  __syncthreads();

  // ---- Phase D: sequential greedy suppression (wave 0) -------------------
  if (tid < 32) {
    uint32_t kw = 0;                     // lane owns keep word `lane`
    for (int i = 0; i < KTOP; ++i) {
      uint32_t row = ioub[i*32 + lane];
      bool sup = (__ballot((kw & row) != 0u) != 0ull);
      bool keepi = (score[i] > CONF_T) && !sup;
      if (keepi && (lane == (i >> 5))) kw |= (1u << (i & 31));
    }
    keepw[lane] = kw;
  }
  __syncthreads();
  if (tid < 32) {                        // exclusive prefix of popcounts
    uint32_t pre = 0;
    for (int t = 0; t < lane; ++t) pre += __popc(keepw[t]);
    pref[lane] = pre;
  }
  __syncthreads();

  // ---- Phase E: emit top-100 kept detections (kept order == score order) -
  float* OB = out + (size_t)b * MAXDET * 6;
  for (int i = tid; i < MAXDET * 6; i += 1024) OB[i] = 0.0f;
  __syncthreads();
  if (tid < KTOP) {
    uint32_t w = keepw[tid >> 5];
    if ((w >> (tid & 31)) & 1u) {
      uint32_t rank = pref[tid >> 5] + __popc(w & ((1u << (tid & 31)) - 1u));
      if (rank < MAXDET) {
        float* o = OB + rank * 6;
        o[0] = rbox[tid*4+0]; o[1] = rbox[tid*4+1];
        o[2] = rbox[tid*4+2]; o[3] = rbox[tid*4+3];
        o[4] = score[tid];    o[5] = clsf[tid];
      }
    }
  }
}

extern "C" void kernel_launch(void* const* d_in, const int* in_sizes, int n_in,
                              void* d_out, int out_size, void* d_ws, size_t ws_size,
                              hipStream_t stream) {
  (void)n_in; (void)out_size; (void)d_ws; (void)ws_size;
  const float* pred = (const float*)d_in[0];
  float* out = (float*)d_out;
  int nb = in_sizes[0] / (NANCH * NCH);
  if (nb < 1) return;
  (void)hipFuncSetAttribute(reinterpret_cast<const void*>(yolo_nms_kernel),
                            hipFuncAttributeMaxDynamicSharedMemorySize,
                            SMEM_BYTES);
  yolo_nms_kernel<<<dim3(nb), dim3(1024), SMEM_BYTES, stream>>>(pred, out);
}